// GRUNetwork_15032385536486
// MI455X (gfx1250) — compile-verified
//
#include <hip/hip_runtime.h>
#include <math.h>

#define B_SZ 16384
#define T_SZ 64

typedef __attribute__((ext_vector_type(2))) float v2f;
typedef __attribute__((ext_vector_type(4))) float v4f;
typedef __attribute__((ext_vector_type(8))) float v8f;

__device__ __forceinline__ v8f wmma4(v2f a, v2f b, v8f c) {
  // D = A(16x4 f32) * B(4x16 f32) + C(16x16 f32), exact fp32 math
  return __builtin_amdgcn_wmma_f32_16x16x4_f32(false, a, false, b, (short)0, c,
                                               false, false);
}

__device__ __forceinline__ float sigmoid_f(float v) {
  return 1.0f / (1.0f + __expf(-v));
}
__device__ __forceinline__ float tanh_f(float x) {
  float ax = fabsf(x);
  float e = __expf(-2.0f * ax);           // e in (0,1], no overflow
  float r = (1.0f - e) / (1.0f + e);
  return copysignf(r, x);
}

// One wave = 16 batch rows of one direction, full T-loop.
// INSZ = 2 (layer 0) or 64 (layers 1,2).
template <int INSZ>
__launch_bounds__(128)
__global__ void gru_layer(const float* __restrict__ xin,   // [B][T][INSZ]
                          const float* __restrict__ Wih,   // [2][96][INSZ]
                          const float* __restrict__ Whh,   // [2][96][32]
                          const float* __restrict__ bih,   // [2][96]
                          const float* __restrict__ bhh,   // [2][96]
                          float* __restrict__ out) {       // [B][T][64]
  constexpr int WPB  = 4;                       // waves per block
  constexpr int WROW = 68;                      // padded LDS row strides
  constexpr int XROW = (INSZ == 64) ? 68 : 2;
  constexpr int HROW = 36;

  __shared__ float wbufp[(INSZ == 64) ? (96 * WROW) : 4];  // Wih (padded)
  __shared__ float xbufp[WPB][16 * XROW];                  // per-wave x tile
  __shared__ float hbufp[WPB][16 * HROW];                  // per-wave h tile

  const int wave = threadIdx.x >> 5;
  const int lane = threadIdx.x & 31;
  const int ln   = lane & 15;
  const int hi   = lane >> 4;                 // 0: lanes 0-15, 1: lanes 16-31
  const int koff = hi ? 2 : 0;

  const int blocksPerDir = (B_SZ / 16) / WPB; // 256
  const int dir  = blockIdx.x / blocksPerDir;
  const int tile = (blockIdx.x % blocksPerDir) * WPB + wave;
  const int b0   = tile * 16;

  const float* WihD = Wih + dir * 96 * INSZ;
  const float* WhhD = Whh + dir * 96 * 32;
  const float* bihD = bih + dir * 96;
  const float* bhhD = bhh + dir * 96;

  // Stage Wih[96][INSZ] into padded LDS (layers 1,2 only)
  if constexpr (INSZ == 64) {
    for (int i = threadIdx.x; i < 96 * 64; i += 128) {
      int g = i >> 6, c = i & 63;
      wbufp[g * WROW + c] = WihD[i];
    }
  }
  __syncthreads();

  // Whh B-fragments held in registers for all 64 steps.
  // B tile tl covers gate cols g = 16*tl + ln; b = (Whh[g][kb], Whh[g][kb+1])
  v2f whhB[6][8];
#pragma unroll
  for (int tl = 0; tl < 6; ++tl)
#pragma unroll
    for (int kc = 0; kc < 8; ++kc)
      whhB[tl][kc] = *(const v2f*)&WhhD[(16 * tl + ln) * 32 + 4 * kc + koff];

  // r/z gates: bih+bhh folded together; n gate keeps them separate
  float brz[4], bxn[2], bhn[2];
  v2f wih0r[6];
#pragma unroll
  for (int tl = 0; tl < 4; ++tl)
    brz[tl] = bihD[16 * tl + ln] + bhhD[16 * tl + ln];
#pragma unroll
  for (int tl = 0; tl < 2; ++tl) {
    bxn[tl] = bihD[16 * (4 + tl) + ln];
    bhn[tl] = bhhD[16 * (4 + tl) + ln];
  }
  if constexpr (INSZ == 2) {
#pragma unroll
    for (int tl = 0; tl < 6; ++tl)
      wih0r[tl] = *(const v2f*)&WihD[(16 * tl + ln) * 2];
  }

  float* xme = xbufp[wave];
  float* hme = hbufp[wave];

  v8f hD[2];      // h in C/D layout (cols 0-15 / 16-31)
  v2f hA[8];      // h in A layout, K-chunks of 4
#pragma unroll
  for (int i = 0; i < 2; ++i)
#pragma unroll
    for (int e = 0; e < 8; ++e) hD[i][e] = 0.0f;
#pragma unroll
  for (int kc = 0; kc < 8; ++kc) { hA[kc].x = 0.0f; hA[kc].y = 0.0f; }

  for (int t = 0; t < T_SZ; ++t) {
    const int tt = dir ? (T_SZ - 1 - t) : t;

    // ---- stage this step's input rows into per-wave LDS ----
    if constexpr (INSZ == 64) {
      const float* src = xin + ((size_t)b0 * T_SZ + tt) * 64;
#pragma unroll
      for (int i = 0; i < 8; ++i) {
        int idx = i * 32 + lane;      // 0..255 float4s
        int row = idx >> 4;           // 0..15
        int col = (idx & 15) * 4;     // 0..60
        v4f v = *(const v4f*)&src[(size_t)row * T_SZ * 64 + col];
        *(v4f*)&xme[row * XROW + col] = v;
      }
    } else {
      int rr = lane >> 1, cc = lane & 1;
      xme[lane] = xin[((size_t)(b0 + rr) * T_SZ + tt) * 2 + cc];
    }

    // A-fragments of x (shared across gate tiles)
    v2f xa[(INSZ == 64) ? 16 : 1];
    if constexpr (INSZ == 64) {
#pragma unroll
      for (int kc = 0; kc < 16; ++kc)
        xa[kc] = *(const v2f*)&xme[ln * XROW + 4 * kc + koff];
    }

    // ---- per 16-column half of h: compute gate triple (r, z, n) ----
#pragma unroll
    for (int tl = 0; tl < 2; ++tl) {
      v8f accR, accZ, accXn, accHn;   // only 4 live accumulators
#pragma unroll
      for (int e = 0; e < 8; ++e) {
        accR[e]  = brz[tl];
        accZ[e]  = brz[2 + tl];
        accXn[e] = bxn[tl];
        accHn[e] = bhn[tl];
      }

      // x @ Wih^T : r and z chain into the fused accumulators
      if constexpr (INSZ == 64) {
#pragma unroll
        for (int kc = 0; kc < 16; ++kc) {
          v2f wr = *(const v2f*)&wbufp[(16 * tl + ln) * WROW + 4 * kc + koff];
          accR = wmma4(xa[kc], wr, accR);
          v2f wz = *(const v2f*)&wbufp[(16 * (2 + tl) + ln) * WROW + 4 * kc + koff];
          accZ = wmma4(xa[kc], wz, accZ);
          v2f wn = *(const v2f*)&wbufp[(16 * (4 + tl) + ln) * WROW + 4 * kc + koff];
          accXn = wmma4(xa[kc], wn, accXn);
        }
      } else {
#pragma unroll
        for (int e = 0; e < 8; ++e) {
          int m = e + 8 * hi;
          float x0 = xme[2 * m], x1 = xme[2 * m + 1];
          accR[e]  += x0 * wih0r[tl].x     + x1 * wih0r[tl].y;
          accZ[e]  += x0 * wih0r[2 + tl].x + x1 * wih0r[2 + tl].y;
          accXn[e] += x0 * wih0r[4 + tl].x + x1 * wih0r[4 + tl].y;
        }
      }

      // h @ Whh^T : r/z fused, n separate (gate needs r * hn)
#pragma unroll
      for (int kc = 0; kc < 8; ++kc) {
        accR  = wmma4(hA[kc], whhB[tl][kc], accR);
        accZ  = wmma4(hA[kc], whhB[2 + tl][kc], accZ);
        accHn = wmma4(hA[kc], whhB[4 + tl][kc], accHn);
      }

      // gates (PyTorch GRU math) + emit h to LDS and global
#pragma unroll
      for (int e = 0; e < 8; ++e) {
        float r = sigmoid_f(accR[e]);
        float z = sigmoid_f(accZ[e]);
        float n = tanh_f(accXn[e] + r * accHn[e]);
        float h2 = (1.0f - z) * n + z * hD[tl][e];
        hD[tl][e] = h2;
        int m = e + 8 * hi;
        int j = 16 * tl + ln;
        hme[m * HROW + j] = h2;
        out[((size_t)(b0 + m) * T_SZ + tt) * 64 + dir * 32 + j] = h2;
      }
    }

    // reload h as A-fragments for the next step
#pragma unroll
    for (int kc = 0; kc < 8; ++kc)
      hA[kc] = *(const v2f*)&hme[ln * HROW + 4 * kc + koff];
  }
}

__global__ void fc_kernel(const float* __restrict__ in,   // [B*T][64]
                          const float* __restrict__ Wfc,  // [2][64]
                          const float* __restrict__ bfc,  // [2]
                          float* __restrict__ out) {      // [B*T][2]
  size_t idx = (size_t)blockIdx.x * blockDim.x + threadIdx.x;
  const float* row = in + idx * 64;
  float a0 = bfc[0], a1 = bfc[1];
#pragma unroll 8
  for (int c = 0; c < 64; ++c) {
    float v = row[c];
    a0 += v * Wfc[c];
    a1 += v * Wfc[64 + c];
  }
  out[idx * 2 + 0] = tanh_f(a0);
  out[idx * 2 + 1] = tanh_f(a1);
}

extern "C" void kernel_launch(void* const* d_in, const int* in_sizes, int n_in,
                              void* d_out, int out_size, void* d_ws,
                              size_t ws_size, hipStream_t stream) {
  (void)in_sizes; (void)n_in; (void)out_size; (void)ws_size;
  const float* x    = (const float*)d_in[0];
  const float* Wih0 = (const float*)d_in[1];
  const float* Whh0 = (const float*)d_in[2];
  const float* bih0 = (const float*)d_in[3];
  const float* bhh0 = (const float*)d_in[4];
  const float* Wih  = (const float*)d_in[5];
  const float* Whh  = (const float*)d_in[6];
  const float* bih  = (const float*)d_in[7];
  const float* bhh  = (const float*)d_in[8];
  const float* Wfc  = (const float*)d_in[9];
  const float* bfc  = (const float*)d_in[10];
  float* out  = (float*)d_out;
  float* buf0 = (float*)d_ws;
  float* buf1 = buf0 + (size_t)B_SZ * T_SZ * 64;

  dim3 blk(128);
  dim3 grd(2 * (B_SZ / 16) / 4);   // 512 blocks: 2 dirs x 1024 tiles / 4 waves

  gru_layer<2><<<grd, blk, 0, stream>>>(x, Wih0, Whh0, bih0, bhh0, buf0);
  gru_layer<64><<<grd, blk, 0, stream>>>(buf0, Wih, Whh, bih, bhh, buf1);
  gru_layer<64><<<grd, blk, 0, stream>>>(buf1, Wih + 2 * 96 * 64,
                                         Whh + 2 * 96 * 32, bih + 2 * 96,
                                         bhh + 2 * 96, buf0);
  fc_kernel<<<(B_SZ * T_SZ) / 256, 256, 0, stream>>>(buf0, Wfc, bfc, out);
}